// SAGE_16552803959273
// MI455X (gfx1250) — compile-verified
//
#include <hip/hip_runtime.h>
#include <hip/hip_bf16.h>
#include <cstdint>

typedef float v2f __attribute__((ext_vector_type(2)));
typedef float v8f __attribute__((ext_vector_type(8)));
typedef int   v4i __attribute__((ext_vector_type(4)));

#define AS1 __attribute__((address_space(1)))
#define AS3 __attribute__((address_space(3)))

#if defined(__has_builtin)
#if __has_builtin(__builtin_amdgcn_global_load_async_to_lds_b128)
#define SAGE_HAS_ASYNC 1
#endif
#endif
#ifndef SAGE_HAS_ASYNC
#define SAGE_HAS_ASYNC 0
#pragma message("gfx1250: __builtin_amdgcn_global_load_async_to_lds_b128 NOT available; using VGPR staging fallback")
#endif

__device__ __forceinline__ void sage_wait_async0() {
#if defined(__has_builtin) && __has_builtin(__builtin_amdgcn_s_wait_asynccnt)
    __builtin_amdgcn_s_wait_asynccnt(0);
#else
    asm volatile("s_wait_asynccnt 0" ::: "memory");
#endif
}

// ---------------------------------------------------------------------------
// degree / inv-degree
// ---------------------------------------------------------------------------
__global__ void sage_degree_kernel(const int* __restrict__ dst,
                                   float* __restrict__ deg, int E) {
    int e = blockIdx.x * blockDim.x + threadIdx.x;
    if (e < E) atomicAdd(&deg[dst[e]], 1.0f);
}

__global__ void sage_invdeg_kernel(float* __restrict__ deg, int N) {
    int i = blockIdx.x * blockDim.x + threadIdx.x;
    if (i < N) {
        float d = deg[i];
        deg[i] = (d > 0.0f) ? (1.0f / d) : 0.0f;
    }
}

// ---------------------------------------------------------------------------
// edge scatter-add: one wave per edge, 128 f32 per edge (float4 per lane).
// agg (51 MB) + feature matrix (51 MB) fit in the 192 MB L2, so the f32
// atomics resolve in L2 rather than HBM.
// ---------------------------------------------------------------------------
__global__ void sage_scatter_add128(const float* __restrict__ P,
                                    const int* __restrict__ src,
                                    const int* __restrict__ dst,
                                    float* __restrict__ agg, int E) {
    int lane = threadIdx.x & 31;
    int edge = blockIdx.x * (blockDim.x >> 5) + (threadIdx.x >> 5);
    if (edge >= E) return;
    int s = src[edge];
    int d = dst[edge];
    const float4 v =
        *reinterpret_cast<const float4*>(P + (size_t)s * 128 + lane * 4);
    float* pd = agg + (size_t)d * 128 + lane * 4;
    atomicAdd(pd + 0, v.x);
    atomicAdd(pd + 1, v.y);
    atomicAdd(pd + 2, v.z);
    atomicAdd(pd + 3, v.w);
}

// ---------------------------------------------------------------------------
// Fused SAGE layer GEMM (fp32 WMMA):
//   C[M,NOUT] = Aself@Wself + (degInv ⊙ Aagg)@Wneigh + bias   (opt. ReLU)
// K = 128. Block = 256 threads (8 waves). Block owns 64 rows; A tile staged
// ONCE per block via GLOBAL_LOAD_ASYNC_TO_LDS_B128 (ASYNCcnt) when available.
// NOUT=128: wave w -> column tile w, 4 row tiles each.
// NOUT=64 : wave w -> column tile w&3, 2 row tiles each (w>>2 selects half).
// inv_deg row-scale applied on A fragments in the neighbor phase (commutes).
// LDS row stride 132 floats -> conflict-free ds_load_b64 fragment reads.
// ---------------------------------------------------------------------------
template <int NOUT, bool RELU>
__global__ __launch_bounds__(256) void sage_gemm_wmma(
    const float* __restrict__ Aself,   // [M,128]
    const float* __restrict__ Aagg,    // [M,128] raw neighbor sums
    const float* __restrict__ degInv,  // [M]
    const float* __restrict__ Wself,   // [128,NOUT] row-major
    const float* __restrict__ Wneigh,  // [128,NOUT] row-major
    const float* __restrict__ bias,    // [NOUT]
    float* __restrict__ C,             // [M,NOUT]
    int M) {
    constexpr int K = 128;
    constexpr int LDSROW = K + 4;
    constexpr int NMT = (NOUT == 128) ? 4 : 2;  // row tiles per wave
    __shared__ float lds[64 * LDSROW];          // ~33.8 KB

    const int tid = threadIdx.x;
    const int wave = tid >> 5;
    const int lane = tid & 31;
    const int m_base = blockIdx.x * 64;
    const int n_tile = (NOUT == 128) ? wave : (wave & 3);
    const int mt0 = (NOUT == 128) ? 0 : ((wave >> 2) * 2);
    const int nc = n_tile * 16 + (lane & 15);   // my output column
    const int koff = (lane >> 4) << 1;          // lanes 16-31: K+2,K+3
    const int arow = lane & 15;

    v8f acc[NMT] = {};

    #pragma unroll
    for (int phase = 0; phase < 2; ++phase) {
        const float* A = phase ? Aagg : Aself;
        const float* W = phase ? Wneigh : Wself;

        if (phase) __syncthreads();             // LDS reuse hazard
        // stage 64x128 f32 A tile, 16B per thread-iteration, coalesced
        for (int f = tid; f < 64 * (K / 4); f += 256) {
            int row = f >> 5;                   // f / (K/4)
            int col4 = f & 31;
            int grow = m_base + row;
            if (grow >= M) grow = M - 1;        // clamp (stores guarded)
            const float* gsrc = A + (size_t)grow * K + col4 * 4;
            float* ldst = &lds[row * LDSROW + col4 * 4];
#if SAGE_HAS_ASYNC
            __builtin_amdgcn_global_load_async_to_lds_b128(
                (AS1 v4i*)gsrc, (AS3 v4i*)ldst, 0, 0);
#else
            *reinterpret_cast<float4*>(ldst) =
                *reinterpret_cast<const float4*>(gsrc);
#endif
        }
#if SAGE_HAS_ASYNC
        sage_wait_async0();
#endif
        __syncthreads();

        // preload all B fragments for K=128 (32 k-steps x v2f = 64 VGPRs)
        v2f bfrag[32];
        #pragma unroll
        for (int ks = 0; ks < 32; ++ks) {
            int kb = ks * 4 + koff;
            v2f t = {W[(size_t)kb * NOUT + nc],
                     W[(size_t)(kb + 1) * NOUT + nc]};
            bfrag[ks] = t;
        }

        #pragma unroll
        for (int mi = 0; mi < NMT; ++mi) {
            const int mt = mt0 + mi;
            float rs = 1.0f;
            if (phase) {                        // per-row inv_deg scale
                int grow = m_base + mt * 16 + arow;
                if (grow >= M) grow = M - 1;
                rs = degInv[grow];
            }
            const float* lp = &lds[(mt * 16 + arow) * LDSROW];
            #pragma unroll
            for (int ks = 0; ks < 32; ++ks) {
                int kb = ks * 4 + koff;
                v2f a = *reinterpret_cast<const v2f*>(lp + kb);  // ds_load_b64
                if (phase) a = a * rs;
                acc[mi] = __builtin_amdgcn_wmma_f32_16x16x4_f32(
                    false, a, false, bfrag[ks], (short)0, acc[mi],
                    false, false);
            }
        }
    }

    const float bv = bias[nc];
    #pragma unroll
    for (int mi = 0; mi < NMT; ++mi) {
        const int mt = mt0 + mi;
        #pragma unroll
        for (int r = 0; r < 8; ++r) {
            int grow = m_base + mt * 16 + ((lane < 16) ? r : (8 + r));
            if (grow < M) {
                float v = acc[mi][r] + bv;
                if (RELU) v = fmaxf(v, 0.0f);
                C[(size_t)grow * NOUT + nc] = v;
            }
        }
    }
}

// ---------------------------------------------------------------------------
// launcher
// ---------------------------------------------------------------------------
extern "C" void kernel_launch(void* const* d_in, const int* in_sizes, int n_in,
                              void* d_out, int out_size, void* d_ws,
                              size_t ws_size, hipStream_t stream) {
    const float* x   = (const float*)d_in[0];
    const int*   src = (const int*)d_in[1];
    const int*   dst = (const int*)d_in[2];
    const float* W1s = (const float*)d_in[3];
    const float* W1n = (const float*)d_in[4];
    const float* b1  = (const float*)d_in[5];
    const float* W2s = (const float*)d_in[6];
    const float* W2n = (const float*)d_in[7];
    const float* b2  = (const float*)d_in[8];
    float* out = (float*)d_out;

    const int N = in_sizes[0] / 128;   // nodes
    const int E = in_sizes[1];         // edges

    // workspace: invdeg [N] | agg [N*128] | h [N*128]  (~103 MB)
    float* deg = (float*)d_ws;
    float* agg = deg + N;
    float* h   = agg + (size_t)N * 128;

    (void)hipMemsetAsync(deg, 0, (size_t)N * sizeof(float), stream);
    (void)hipMemsetAsync(agg, 0, (size_t)N * 128 * sizeof(float), stream);

    sage_degree_kernel<<<(E + 255) / 256, 256, 0, stream>>>(dst, deg, E);
    sage_scatter_add128<<<(E + 7) / 8, 256, 0, stream>>>(x, src, dst, agg, E);
    sage_invdeg_kernel<<<(N + 255) / 256, 256, 0, stream>>>(deg, N);

    // layer 1: h = relu(x@W1s + (inv ⊙ agg)@W1n + b1)
    sage_gemm_wmma<128, true>
        <<<dim3((N + 63) / 64, 1), 256, 0, stream>>>(x, agg, deg, W1s, W1n,
                                                     b1, h, N);

    // layer 2 aggregation of h
    (void)hipMemsetAsync(agg, 0, (size_t)N * 128 * sizeof(float), stream);
    sage_scatter_add128<<<(E + 7) / 8, 256, 0, stream>>>(h, src, dst, agg, E);

    // layer 2: out = h@W2s + (inv ⊙ agg)@W2n + b2
    sage_gemm_wmma<64, false>
        <<<dim3((N + 63) / 64, 1), 256, 0, stream>>>(h, agg, deg, W2s, W2n,
                                                     b2, out, N);
}